// gMamba_61950608277761
// MI455X (gfx1250) — compile-verified
//
#include <hip/hip_runtime.h>
#include <cstddef>

// ---------------- problem constants ----------------
#define L_SEQ   65536
#define DMODEL  12
#define DINNER  24
#define DSTATE  16
#define C_IN    64
#define C_MID   32
#define HGRU    6
#define MROWS   (L_SEQ * DMODEL)     // 786432 GEMM rows for conv1
#define NC      1024                 // scan chunks
#define CL      (L_SEQ / NC)         // 64 steps per chunk

typedef float v2f __attribute__((ext_vector_type(2)));
typedef float v8f __attribute__((ext_vector_type(8)));

// ---------------- workspace layout (float offsets) ----------------
#define OFF_SEQ   0
#define OFF_U     (OFF_SEQ   + MROWS)
#define OFF_DELTA (OFF_U     + L_SEQ * DINNER)
#define OFF_B     (OFF_DELTA + L_SEQ * DINNER)
#define OFF_C     (OFF_B     + L_SEQ * DSTATE)
#define OFF_GIF   (OFF_C     + L_SEQ * DSTATE)
#define OFF_GIB   (OFF_GIF   + L_SEQ * 18)
#define OFF_HC    (OFF_GIB   + L_SEQ * 18)
#define OFF_SD    (OFF_HC    + NC * DINNER * DSTATE)
#define OFF_YL    (OFF_SD    + NC * DINNER)
#define OFF_HF    (OFF_YL    + DINNER)
#define OFF_HB    (OFF_HF    + HGRU)
#define OFF_YB0   (OFF_HB    + HGRU)

// =================================================================
// K1: fused conv1(64->32,1x1)+bn1+relu+conv2(32->1,1x1)+bn2+relu.
// conv1 as GEMM via V_WMMA_F32_16X16X4_F32. A[m][k]=x[k*MROWS+m].
// Wave tile = 16 rows x 32 cols (two 16x16 accumulators), K-loop 16x.
// C layout (ISA 7.12.2): lanes 0-15 vgpr r = D[m0+r][n0+lane],
// lanes 16-31 vgpr r = D[m0+8+r][n0+lane-16].
// =================================================================
__global__ __launch_bounds__(256) void k1_conv(
    const float* __restrict__ x,
    const float* __restrict__ w1, const float* __restrict__ cb1,
    const float* __restrict__ g1, const float* __restrict__ b1,
    const float* __restrict__ m1, const float* __restrict__ v1,
    const float* __restrict__ w2, const float* __restrict__ cb2,
    const float* __restrict__ g2, const float* __restrict__ b2,
    const float* __restrict__ m2, const float* __restrict__ v2,
    float* __restrict__ seq)
{
  __shared__ float sW[C_MID * C_IN];
  for (int i = threadIdx.x; i < C_MID * C_IN; i += 256) sW[i] = w1[i];
  __syncthreads();

  const int lane = threadIdx.x & 31;
  const int wv   = threadIdx.x >> 5;
  const int l16  = lane & 15;
  const int half = lane >> 4;                 // which K pair / which M half
  const int m0   = (blockIdx.x * 8 + wv) * 16;

  const float* xA = x + (size_t)(half * 2) * MROWS + m0 + l16;

  v8f c0 = {};
  v8f c1 = {};
#pragma unroll
  for (int kk = 0; kk < 16; ++kk) {
    const int k0 = kk * 4;
    v2f a, bA, bB;
    // A 16x4 f32 frag: lanes 0-15 hold K=k0,k0+1; lanes 16-31 hold K=k0+2,k0+3
    a.x = xA[(size_t)k0 * MROWS];
    a.y = xA[(size_t)(k0 + 1) * MROWS];
    if (kk < 15) __builtin_prefetch(&xA[(size_t)(k0 + 4) * MROWS], 0, 1);
    // B 4x16 f32 frag (mirrored layout): B[k][n] = w1[n*64+k]
    const int kb = k0 + half * 2;
    bA.x = sW[l16 * C_IN + kb];
    bA.y = sW[l16 * C_IN + kb + 1];
    bB.x = sW[(16 + l16) * C_IN + kb];
    bB.y = sW[(16 + l16) * C_IN + kb + 1];
    c0 = __builtin_amdgcn_wmma_f32_16x16x4_f32(false, a, false, bA, (short)0, c0, false, false);
    c1 = __builtin_amdgcn_wmma_f32_16x16x4_f32(false, a, false, bB, (short)0, c1, false, false);
  }

  // bn1 + relu per column, then conv2 dot over 32 channels via lane reduction
  const int nA = l16, nB = 16 + l16;
  const float sA = g1[nA] * rsqrtf(v1[nA] + 1e-5f);
  const float tA = cb1[nA] * sA + b1[nA] - m1[nA] * sA;
  const float sB = g1[nB] * rsqrtf(v1[nB] + 1e-5f);
  const float tB = cb1[nB] * sB + b1[nB] - m1[nB] * sB;
  const float w2A = w2[nA], w2B = w2[nB];

  float p[8];
#pragma unroll
  for (int r = 0; r < 8; ++r) {
    float vA = c0[r] * sA + tA; vA = vA > 0.f ? vA : 0.f;
    float vB = c1[r] * sB + tB; vB = vB > 0.f ? vB : 0.f;
    p[r] = vA * w2A + vB * w2B;
  }
#pragma unroll
  for (int off = 8; off >= 1; off >>= 1) {
#pragma unroll
    for (int r = 0; r < 8; ++r) p[r] += __shfl_xor(p[r], off, 32);
  }
  if (l16 == 0) {
    const float s2 = g2[0] * rsqrtf(v2[0] + 1e-5f);
    const float t2 = cb2[0] * s2 + b2[0] - m2[0] * s2;
#pragma unroll
    for (int r = 0; r < 8; ++r) {
      float v = p[r] * s2 + t2;
      seq[m0 + half * 8 + r] = v > 0.f ? v : 0.f;
    }
  }
}

// =================================================================
// K2a: per-t parallel part of Mamba + GRU input projections.
// in_proj -> xm; causal depthwise conv1d (k=4) via LDS halo; silu;
// x_proj -> dt/B/C; delta=softplus(dt*wdt+bdt); gi_fwd/gi_bwd GEMVs.
// =================================================================
__global__ __launch_bounds__(256) void k2a_proj(
    const float* __restrict__ seq,
    const float* __restrict__ Wip, const float* __restrict__ w1d,
    const float* __restrict__ b1d, const float* __restrict__ Wxp,
    const float* __restrict__ wdt, const float* __restrict__ bdt,
    const float* __restrict__ Wih, const float* __restrict__ bih,
    float* __restrict__ u, float* __restrict__ delta,
    float* __restrict__ Bv, float* __restrict__ Cv,
    float* __restrict__ gif, float* __restrict__ gib)
{
  __shared__ float sxm[256 + 3][DINNER];
  const int tid = threadIdx.x;
  const int t = blockIdx.x * 256 + tid;

  float sv[DMODEL];
#pragma unroll
  for (int w = 0; w < DMODEL; ++w) sv[w] = seq[t * DMODEL + w];

  for (int j = 0; j < DINNER; ++j) {
    float acc = 0.f;
#pragma unroll
    for (int w = 0; w < DMODEL; ++w) acc += Wip[j * DMODEL + w] * sv[w];
    sxm[tid + 3][j] = acc;
  }
  if (tid < 3) {                      // left halo (zeros for causal pad t<0)
    const int th = blockIdx.x * 256 - 3 + tid;
    for (int j = 0; j < DINNER; ++j) {
      float acc = 0.f;
      if (th >= 0)
        for (int w = 0; w < DMODEL; ++w) acc += Wip[j * DMODEL + w] * seq[th * DMODEL + w];
      sxm[tid][j] = acc;
    }
  }
  __syncthreads();

  float uu[DINNER];
  for (int j = 0; j < DINNER; ++j) {
    float xc = b1d[j];
#pragma unroll
    for (int k = 0; k < 4; ++k) xc += sxm[tid + k][j] * w1d[j * 4 + k];
    uu[j] = xc / (1.f + expf(-xc));   // silu
    u[t * DINNER + j] = uu[j];
  }

  float dt = 0.f;
  for (int j = 0; j < DINNER; ++j) dt += Wxp[j] * uu[j];
  for (int n = 0; n < DSTATE; ++n) {
    float b = 0.f, c = 0.f;
#pragma unroll
    for (int j = 0; j < DINNER; ++j) {
      b += Wxp[(1 + n) * DINNER + j] * uu[j];
      c += Wxp[(1 + DSTATE + n) * DINNER + j] * uu[j];
    }
    Bv[t * DSTATE + n] = b;
    Cv[t * DSTATE + n] = c;
  }
  for (int j = 0; j < DINNER; ++j) {
    const float z = dt * wdt[j] + bdt[j];
    delta[t * DINNER + j] = (z > 20.f) ? z : log1pf(expf(z));
  }
  for (int g = 0; g < 18; ++g) {
    float af = bih[g], ab = bih[18 + g];
#pragma unroll
    for (int w = 0; w < DMODEL; ++w) {
      af += Wih[g * DMODEL + w] * sv[w];
      ab += Wih[(18 + g) * DMODEL + w] * sv[w];
    }
    gif[(size_t)t * 18 + g] = af;
    gib[(size_t)(L_SEQ - 1 - t) * 18 + g] = ab;   // store reversed for bwd GRU
  }
}

// =================================================================
// K2b: chunked linear selective scan. One (d,n) per thread, 64 steps.
// Emits partial state Hc (h with h0=0) and per-d sum of delta (SD).
// =================================================================
__global__ __launch_bounds__(384) void k2b_chunk(
    const float* __restrict__ A_log,
    const float* __restrict__ u, const float* __restrict__ delta,
    const float* __restrict__ Bv,
    float* __restrict__ Hc, float* __restrict__ SD)
{
  const int tid = threadIdx.x;            // 0..383, tid = d*16+n
  const int d = tid >> 4, n = tid & 15;
  const float Adn = -expf(A_log[tid]);
  const int t0 = blockIdx.x * CL;
  float h = 0.f, sd = 0.f;
  for (int i = 0; i < CL; ++i) {
    const int t = t0 + i;
    const float dl = delta[t * DINNER + d];
    h = h * expf(dl * Adn) + dl * u[t * DINNER + d] * Bv[t * DSTATE + n];
    sd += dl;
  }
  Hc[blockIdx.x * 384 + tid] = h;
  if (n == 0) SD[blockIdx.x * DINNER + d] = sd;
}

// =================================================================
// K3: sequential combine of NC chunks: h = exp(A*SDc)*h + Hc.
// Then y[L-1][d] = sum_n h*C[L-1][n] + u[L-1][d]*Dp[d].
// =================================================================
__global__ __launch_bounds__(384) void k3_combine(
    const float* __restrict__ A_log, const float* __restrict__ Dp,
    const float* __restrict__ Hc, const float* __restrict__ SD,
    const float* __restrict__ u, const float* __restrict__ Cv,
    float* __restrict__ yl)
{
  const int tid = threadIdx.x;
  const int d = tid >> 4, n = tid & 15;
  const float Adn = -expf(A_log[tid]);
  float h = 0.f;
  for (int c = 0; c < NC; ++c)
    h = h * expf(Adn * SD[c * DINNER + d]) + Hc[c * 384 + tid];
  float val = h * Cv[(size_t)(L_SEQ - 1) * DSTATE + n];
#pragma unroll
  for (int off = 8; off >= 1; off >>= 1) val += __shfl_xor(val, off, 32);
  if (n == 0) yl[d] = val + u[(size_t)(L_SEQ - 1) * DINNER + d] * Dp[d];
}

// =================================================================
// K4: the two sequential GRUs. Wave 0 = forward, wave 1 = backward.
// Lanes 0-5 = r gates, 6-11 = z, 12-17 = n; h broadcast via shuffles.
// gi precomputed by K2a (bwd stored time-reversed).
// =================================================================
__global__ __launch_bounds__(64) void k4_gru(
    const float* __restrict__ hidden,
    const float* __restrict__ Whh, const float* __restrict__ bhh,
    const float* __restrict__ gif, const float* __restrict__ gib,
    float* __restrict__ hf, float* __restrict__ hb, float* __restrict__ yb0)
{
  const int lane = threadIdx.x & 31;
  const int dir  = threadIdx.x >> 5;
  const float* __restrict__ gi = dir ? gib : gif;
  const int j = lane < 18 ? lane : 17;

  const float wh0 = Whh[dir * 108 + j * 6 + 0];
  const float wh1 = Whh[dir * 108 + j * 6 + 1];
  const float wh2 = Whh[dir * 108 + j * 6 + 2];
  const float wh3 = Whh[dir * 108 + j * 6 + 3];
  const float wh4 = Whh[dir * 108 + j * 6 + 4];
  const float wh5 = Whh[dir * 108 + j * 6 + 5];
  const float bh  = bhh[dir * 18 + j];

  float hv = hidden[dir * HGRU + (lane < HGRU ? lane : 0)];
  float h0 = __shfl(hv, 0, 32), h1 = __shfl(hv, 1, 32), h2 = __shfl(hv, 2, 32);
  float h3 = __shfl(hv, 3, 32), h4 = __shfl(hv, 4, 32), h5 = __shfl(hv, 5, 32);

  for (int t = 0; t < L_SEQ; ++t) {
    const float g  = gi[(size_t)t * 18 + j];
    const float gh = bh + wh0 * h0 + wh1 * h1 + wh2 * h2 + wh3 * h3 + wh4 * h4 + wh5 * h5;
    const float sg = 1.f / (1.f + expf(-(g + gh)));     // r (lanes 0-5), z (6-11)
    const float r  = __shfl(sg, (lane - 12) & 31, 32);  // lanes 12-17 <- 0-5
    const float nv = tanhf(g + r * gh);                 // n gates
    const float z  = __shfl(sg, (lane + 6) & 31, 32);   // lanes 0-5 <- 6-11
    const float nn = __shfl(nv, (lane + 12) & 31, 32);  // lanes 0-5 <- 12-17
    hv = (1.f - z) * nn + z * hv;                       // valid in lanes 0-5
    h0 = __shfl(hv, 0, 32); h1 = __shfl(hv, 1, 32); h2 = __shfl(hv, 2, 32);
    h3 = __shfl(hv, 3, 32); h4 = __shfl(hv, 4, 32); h5 = __shfl(hv, 5, 32);
    if (t == 0 && dir == 1 && lane < HGRU) yb0[lane] = hv;  // yb[:,0]
  }
  if (lane < HGRU) { if (dir) hb[lane] = hv; else hf[lane] = hv; }
}

// =================================================================
// K5: final 13 outputs: pred = (out1[L-1]+out2[L-1]) @ lin_w + lin_b,
// then hf, hb. out2[L-1] = concat(hf, yb0).
// =================================================================
__global__ void k5_final(
    const float* __restrict__ seq, const float* __restrict__ Wip,
    const float* __restrict__ Wout, const float* __restrict__ lw,
    const float* __restrict__ lb, const float* __restrict__ yl,
    const float* __restrict__ hf, const float* __restrict__ hb,
    const float* __restrict__ yb0, float* __restrict__ out)
{
  if (threadIdx.x != 0 || blockIdx.x != 0) return;
  float sv[DMODEL];
  for (int w = 0; w < DMODEL; ++w) sv[w] = seq[(size_t)(L_SEQ - 1) * DMODEL + w];
  float o1[DMODEL];
  for (int w = 0; w < DMODEL; ++w) o1[w] = 0.f;
  for (int d = 0; d < DINNER; ++d) {
    float res = 0.f;
    for (int w = 0; w < DMODEL; ++w) res += Wip[(DINNER + d) * DMODEL + w] * sv[w];
    const float g = yl[d] * (res / (1.f + expf(-res)));   // y * silu(res)
    for (int w = 0; w < DMODEL; ++w) o1[w] += g * Wout[w * DINNER + d];
  }
  float pred = lb[0];
  for (int w = 0; w < DMODEL; ++w) {
    const float o2 = (w < HGRU) ? hf[w] : yb0[w - HGRU];
    pred += (o1[w] + o2) * lw[w];
  }
  out[0] = pred;
  for (int k = 0; k < HGRU; ++k) { out[1 + k] = hf[k]; out[7 + k] = hb[k]; }
}

// =================================================================
extern "C" void kernel_launch(void* const* d_in, const int* in_sizes, int n_in,
                              void* d_out, int out_size, void* d_ws, size_t ws_size,
                              hipStream_t stream)
{
  (void)in_sizes; (void)n_in; (void)out_size; (void)ws_size;
  const float* x         = (const float*)d_in[0];
  const float* hidden    = (const float*)d_in[1];
  const float* conv1_w   = (const float*)d_in[2];
  const float* conv1_b   = (const float*)d_in[3];
  const float* bn1_g     = (const float*)d_in[4];
  const float* bn1_b     = (const float*)d_in[5];
  const float* bn1_m     = (const float*)d_in[6];
  const float* bn1_v     = (const float*)d_in[7];
  const float* conv2_w   = (const float*)d_in[8];
  const float* conv2_b   = (const float*)d_in[9];
  const float* bn2_g     = (const float*)d_in[10];
  const float* bn2_b     = (const float*)d_in[11];
  const float* bn2_m     = (const float*)d_in[12];
  const float* bn2_v     = (const float*)d_in[13];
  const float* in_proj_w = (const float*)d_in[14];
  const float* conv1d_w  = (const float*)d_in[15];
  const float* conv1d_b  = (const float*)d_in[16];
  const float* x_proj_w  = (const float*)d_in[17];
  const float* dt_proj_w = (const float*)d_in[18];
  const float* dt_proj_b = (const float*)d_in[19];
  const float* A_log     = (const float*)d_in[20];
  const float* Dp        = (const float*)d_in[21];
  const float* out_proj_w= (const float*)d_in[22];
  const float* gru_Wih   = (const float*)d_in[23];
  const float* gru_Whh   = (const float*)d_in[24];
  const float* gru_bih   = (const float*)d_in[25];
  const float* gru_bhh   = (const float*)d_in[26];
  const float* lin_w     = (const float*)d_in[27];
  const float* lin_b     = (const float*)d_in[28];

  float* ws    = (float*)d_ws;
  float* seq   = ws + OFF_SEQ;
  float* u     = ws + OFF_U;
  float* delta = ws + OFF_DELTA;
  float* Bv    = ws + OFF_B;
  float* Cv    = ws + OFF_C;
  float* gif   = ws + OFF_GIF;
  float* gib   = ws + OFF_GIB;
  float* Hc    = ws + OFF_HC;
  float* SD    = ws + OFF_SD;
  float* yl    = ws + OFF_YL;
  float* hf    = ws + OFF_HF;
  float* hb    = ws + OFF_HB;
  float* yb0   = ws + OFF_YB0;

  k1_conv<<<MROWS / 128, 256, 0, stream>>>(x, conv1_w, conv1_b, bn1_g, bn1_b, bn1_m, bn1_v,
                                           conv2_w, conv2_b, bn2_g, bn2_b, bn2_m, bn2_v, seq);
  k2a_proj<<<L_SEQ / 256, 256, 0, stream>>>(seq, in_proj_w, conv1d_w, conv1d_b, x_proj_w,
                                            dt_proj_w, dt_proj_b, gru_Wih, gru_bih,
                                            u, delta, Bv, Cv, gif, gib);
  k2b_chunk<<<NC, 384, 0, stream>>>(A_log, u, delta, Bv, Hc, SD);
  k4_gru<<<1, 64, 0, stream>>>(hidden, gru_Whh, gru_bhh, gif, gib, hf, hb, yb0);
  k3_combine<<<1, 384, 0, stream>>>(A_log, Dp, Hc, SD, u, Cv, yl);
  k5_final<<<1, 32, 0, stream>>>(seq, in_proj_w, out_proj_w, lin_w, lin_b,
                                 yl, hf, hb, yb0, (float*)d_out);
}